// SoftDTW_43739946943560
// MI455X (gfx1250) — compile-verified
//
#include <hip/hip_runtime.h>

// SoftDTW-like row recurrence over T=1024 with left-neighbor coupling over D=2048.
// 128 workgroups = 32 batches x 4 column-slices, 768 threads (24 wave32).
// Per-wave: 64 owned cols (2/lane) + 32-col halo (1/lane), ds_bpermute neighbor
// exchange, LDS carry exchange + barrier every 32 steps, async global->LDS
// prefetch of x rows (PF deep) gated by s_wait_asynccnt.

#define T_DIM 1024
#define D_DIM 2048
#define NB 12   // LDS staging slots
#define PF 10   // prefetch depth in rows (< NB)

#if defined(__HIP_DEVICE_COMPILE__)
#if __has_builtin(__builtin_amdgcn_global_load_async_to_lds_b64) && \
    __has_builtin(__builtin_amdgcn_global_load_async_to_lds_b32)
#define ASYNC_BUILTIN 1
#warning "CDNA5 async path: __builtin_amdgcn_global_load_async_to_lds_*"
#else
#define ASYNC_BUILTIN 0
#warning "CDNA5 async path: inline-asm global_load_async_to_lds_*"
#endif
#else
#define ASYNC_BUILTIN 0
#endif

typedef float v2f __attribute__((ext_vector_type(2)));
typedef int   v2i __attribute__((vector_size(8)));   // matches builtin prototype

typedef __attribute__((address_space(1))) v2i as1_v2i;  // global, b64 element
typedef __attribute__((address_space(3))) v2i as3_v2i;  // LDS,    b64 element
typedef __attribute__((address_space(1))) int as1_i32;  // global, b32 element
typedef __attribute__((address_space(3))) int as3_i32;  // LDS,    b32 element

__device__ __forceinline__ float softmix(float a, float b) {
  // b + z*sigmoid(z), z=a-b ; sigmoid via exp2/rcp (hw trans, ~1ulp)
  float z = a - b;
  float e = __builtin_amdgcn_exp2f(z * -1.44269504088896340736f); // exp(-z)
  float p = __builtin_amdgcn_rcpf(1.0f + e);
  return __builtin_fmaf(z, p, b);
}

__device__ __forceinline__ v2f softmix2(v2f a, v2f b) {
  // packed-f32 version: sub/mul/add/fma lower to VOP3P v_pk_*_f32
  v2f z = a - b;
  v2f arg = z * -1.44269504088896340736f;
  v2f e = { __builtin_amdgcn_exp2f(arg.x), __builtin_amdgcn_exp2f(arg.y) };
  v2f den = e + 1.0f;
  v2f p = { __builtin_amdgcn_rcpf(den.x), __builtin_amdgcn_rcpf(den.y) };
  return z * p + b;
}

__device__ __forceinline__ void async_b64(const float* xb, int goff, unsigned ldsb) {
#if ASYNC_BUILTIN
  __builtin_amdgcn_global_load_async_to_lds_b64(
      (as1_v2i*)((const char*)xb + goff),
      (as3_v2i*)(unsigned long long)ldsb, 0, 0);
#else
  // GVS: mem = SADDR(u64) + VADDR(i32) ; LDS dest byte offset in first operand
  asm volatile("global_load_async_to_lds_b64 %0, %1, %2"
               :: "v"(ldsb), "v"(goff), "s"(xb) : "memory");
#endif
}

__device__ __forceinline__ void async_b32(const float* xb, int goff, unsigned ldsb) {
#if ASYNC_BUILTIN
  __builtin_amdgcn_global_load_async_to_lds_b32(
      (as1_i32*)((const char*)xb + goff),
      (as3_i32*)(unsigned long long)ldsb, 0, 0);
#else
  asm volatile("global_load_async_to_lds_b32 %0, %1, %2"
               :: "v"(ldsb), "v"(goff), "s"(xb) : "memory");
#endif
}

extern "C" __global__ __launch_bounds__(768, 1)
void softscan_kernel(const float* __restrict__ x, float* __restrict__ out) {
  extern __shared__ float smem[];
  // layout: [NB][24 waves][96 cols] staging ; then [2][2048] carry exchange

  const int blk = blockIdx.x;
  const int b   = blk >> 2;     // batch
  const int s   = blk & 3;      // column slice
  const int tid = threadIdx.x;
  const int w   = tid >> 5;     // wave in workgroup (0..23)
  const int L   = tid & 31;     // lane
  const int B   = 512 * s - 1024 + 64 * w;  // wave's first owned column
  if (B < 0) return;            // outside the cone (whole waves)

  const float* xb = x   + (size_t)b * (T_DIM * D_DIM);
  float*       ob = out + (size_t)b * (T_DIM * D_DIM);

  const int dA  = B + 2 * L;            // owned col pair dA, dA+1
  const int dH  = B - 32 + L;           // halo col
  const int dHc = dH < 0 ? 0 : dH;
  const int g64 = B - 32 + 2 * L;       // staged b64 cols (halo+lower owned)
  const int g64c = g64 < 0 ? 0 : g64;   // clamp keeps 8B alignment (even col)
  const int g32 = B + 32 + L;           // staged b32 col (upper owned)
  const bool owner = (w >= 16);         // owns cols [512s, 512s+512)
  const int Tlast  = B - 512 * s + 1086; // last step this wave's cols are needed

  float* xch = smem + NB * 24 * 96;
  const unsigned stg_base = (unsigned)(size_t)smem;  // LDS byte offset of staging
  const unsigned l64b = stg_base + (unsigned)(w * 96 + 2 * L) * 4u;
  const unsigned l32b = stg_base + (unsigned)(w * 96 + 64 + L) * 4u;
  const int c64 = g64c * 4;
  const int c32 = g32 * 4;

  // state t=0 is x row 0; halo init is exact where it exists
  float cA, cB, h;
  {
    v2f c0 = *(const v2f*)(xb + dA);
    cA = c0.x; cB = c0.y;
    h = xb[dHc];
    if (owner) *(v2f*)(ob + dA) = c0;    // out row 0 = x row 0
  }

  // prologue: stage x rows 1..PF
  for (int r = 1; r <= PF; ++r) {
    unsigned lofs = (unsigned)r * (24u * 96u * 4u);
    int go = r * (D_DIM * 4);
    async_b64(xb, go + c64, l64b + lofs);
    async_b32(xb, go + c32, l32b + lofs);
  }

  const int rot = ((L + 31) & 31) << 2;  // bpermute byte index: lane L <- lane L-1 (wrap)
  int slot_c = 1;
  int slot_p = 1 + PF;
  int buf = 0;

  for (int t = 1; t < T_DIM; ++t) {
    // halo refresh every 32 steps (state t-1 is a multiple of 32)
    if (t > 1 && (t & 31) == 1) {
      float* xc = xch + (buf << 11);
      *(v2f*)(xc + dA) = (v2f){cA, cB};
      __syncthreads();
      h = xc[dHc];
      buf ^= 1;
      if (Tlast <= t - 1) return;        // cone passed this wave: retire
    }

    // prefetch row t+PF into private LDS slot
    int tp = t + PF;
    if (tp < T_DIM) {
      unsigned lofs = (unsigned)slot_p * (24u * 96u * 4u);
      int go = tp * (D_DIM * 4);
      async_b64(xb, go + c64, l64b + lofs);
      async_b32(xb, go + c32, l32b + lofs);
    }
    // oldest in-flight row (t) complete when <= 2*PF async ops outstanding
    asm volatile("s_wait_asynccnt %0" :: "n"(2 * PF) : "memory");

    const float* row = smem + slot_c * (24 * 96) + w * 96;
    float xH  = row[L];
    v2f   xab = *(const v2f*)(row + 32 + 2 * L);

    // left neighbors via wave rotation
    float hl = __int_as_float(__builtin_amdgcn_ds_bpermute(rot, __float_as_int(h)));
    float cl = __int_as_float(__builtin_amdgcn_ds_bpermute(rot, __float_as_int(cB)));
    float left0 = (L == 0) ? hl : cl;    // lane0's left owned neighbor is halo[31]

    float nH = xH + softmix(h, hl);      // halo advance (scalar)
    v2f a2 = {cA, cB};
    v2f b2 = {left0, cA};
    v2f s2 = softmix2(a2, b2);           // owned pair (packed f32)
    float sx = (dA == 0) ? cA : s2.x;    // col 0 anchored
    v2f n2 = xab + (v2f){sx, s2.y};

    if (owner)
      *(v2f*)(ob + (size_t)t * D_DIM + dA) = n2;

    h = nH; cA = n2.x; cB = n2.y;
    if (++slot_c == NB) slot_c = 0;
    if (++slot_p == NB) slot_p = 0;
  }
}

extern "C" void kernel_launch(void* const* d_in, const int* in_sizes, int n_in,
                              void* d_out, int out_size, void* d_ws, size_t ws_size,
                              hipStream_t stream) {
  const float* x = (const float*)d_in[0];
  float* out = (float*)d_out;
  size_t shmem = (size_t)(NB * 24 * 96 + 2 * 2048) * sizeof(float); // 124 KB
  hipLaunchKernelGGL(softscan_kernel, dim3(128), dim3(768), shmem, stream, x, out);
}